// RPN_3D_loss_17738214932997
// MI455X (gfx1250) — compile-verified
//
#include <hip/hip_runtime.h>

#define AS1 __attribute__((address_space(1)))
#define AS3 __attribute__((address_space(3)))

constexpr int Bb  = 4;
constexpr int Nn  = 110592;
constexpr int Gg  = 64;
constexpr int TPB = 256;
constexpr int BLK_X = Nn / TPB;     // 432 (exact)
constexpr int NBLK  = BLK_X * Bb;   // 1728 partial rows

#if __has_builtin(__builtin_amdgcn_global_load_async_to_lds_b32)
#define HAVE_ASYNC_G2L 1
#endif

// Copy one dword global -> LDS. On gfx1250 toolchains with the async builtin this
// emits global_load_async_to_lds_b32 (tracked by ASYNCcnt); otherwise a plain
// load+ds_store fallback.
__device__ __forceinline__ void g2l_b32(const void* gsrc, void* ldst) {
#ifdef HAVE_ASYNC_G2L
  __builtin_amdgcn_global_load_async_to_lds_b32(
      (AS1 int*)(unsigned long long)gsrc,
      (AS3 int*)(unsigned)(unsigned long long)ldst,
      0, 0);
#else
  *(unsigned*)ldst = *(const unsigned*)gsrc;
#endif
}

__device__ __forceinline__ void stage_wait() {
#ifdef HAVE_ASYNC_G2L
 #if __has_builtin(__builtin_amdgcn_s_wait_asynccnt)
  __builtin_amdgcn_s_wait_asynccnt(0);
 #else
  asm volatile("s_wait_asynccnt 0" ::: "memory");
 #endif
#endif
  __syncthreads();
}

__global__ __launch_bounds__(TPB) void rpn_loss_main(
    const float* __restrict__ cls,    // [B,N,4]
    const float* __restrict__ b2,     // [B,N,4]
    const float* __restrict__ b3,     // [B,N,7]
    const float* __restrict__ rois,   // [B,N,5]
    const float* __restrict__ r3,     // [B,N,9]
    const float* __restrict__ g2,     // [B,G,4]
    const float* __restrict__ g3,     // [B,G,7]
    const int*   __restrict__ gc,     // [B,G]
    const float* __restrict__ means,  // [1,13]
    const float* __restrict__ stds,   // [1,13]
    float* __restrict__ partials)     // [NBLK*4]
{
  __shared__ float  s_g2d[Gg * 4];
  __shared__ float  s_g3d[Gg * 7];
  __shared__ int    s_gc[Gg];
  __shared__ float4 s_red[TPB];

  const int tid = threadIdx.x;
  const int b   = blockIdx.y;
  const int n   = blockIdx.x * TPB + tid;

  // --- stage per-batch GT tables into LDS (async on gfx1250) ---
  {
    const float* g2b = g2 + (size_t)b * Gg * 4;
    const float* g3b = g3 + (size_t)b * Gg * 7;
    const int*   gcb = gc + (size_t)b * Gg;
    for (int i = tid; i < Gg * 4; i += TPB) g2l_b32(g2b + i, s_g2d + i);
    for (int i = tid; i < Gg * 7; i += TPB) g2l_b32(g3b + i, s_g3d + i);
    for (int i = tid; i < Gg;     i += TPB) g2l_b32(gcb + i, s_gc  + i);
  }
  stage_wait();

  const size_t bn = (size_t)b * Nn + n;

  // --- anchor geometry ---
  const float ax1 = rois[bn * 5 + 0];
  const float ay1 = rois[bn * 5 + 1];
  const float ax2 = rois[bn * 5 + 2];
  const float ay2 = rois[bn * 5 + 3];
  const float aw = ax2 - ax1 + 1.0f;
  const float ah = ay2 - ay1 + 1.0f;
  const float area_a = aw * ah;

  // --- IoU argmax over G (LDS broadcast reads) ---
  float best_iou = -1.0f;
  int best = 0;
#pragma unroll 4
  for (int g = 0; g < Gg; ++g) {
    const float gx1 = s_g2d[g * 4 + 0];
    const float gy1 = s_g2d[g * 4 + 1];
    const float gx2 = s_g2d[g * 4 + 2];
    const float gy2 = s_g2d[g * 4 + 3];
    const float iw = fminf(ax2, gx2) - fmaxf(ax1, gx1) + 1.0f;
    const float ih = fminf(ay2, gy2) - fmaxf(ay1, gy1) + 1.0f;
    const float inter = fmaxf(iw, 0.0f) * fmaxf(ih, 0.0f);
    const float area_g = (gx2 - gx1 + 1.0f) * (gy2 - gy1 + 1.0f);
    const float iou = inter / (area_a + area_g - inter);
    if (iou > best_iou) { best_iou = iou; best = g; }  // first-max wins (jnp.argmax)
  }
  const bool  is_fg = best_iou >= 0.5f;
  const float fg    = is_fg ? 1.0f : 0.0f;
  const int   label = is_fg ? s_gc[best] : 0;

  // --- target encoding ---
  const float acx = ax1 + 0.5f * aw;
  const float acy = ay1 + 0.5f * ah;

  const float mx1 = s_g2d[best * 4 + 0];
  const float my1 = s_g2d[best * 4 + 1];
  const float mx2 = s_g2d[best * 4 + 2];
  const float my2 = s_g2d[best * 4 + 3];
  const float gw  = mx2 - mx1 + 1.0f;
  const float gh  = my2 - my1 + 1.0f;
  const float gcx = mx1 + 0.5f * gw;
  const float gcy = my1 + 0.5f * gh;

  const float z0  = r3[bn * 9 + 4];
  const float w0  = r3[bn * 9 + 5];
  const float h0  = r3[bn * 9 + 6];
  const float l0  = r3[bn * 9 + 7];
  const float ry0 = r3[bn * 9 + 8];
  const float* m3 = s_g3d + best * 7;

  float t[11];
  t[0]  = (gcx - acx) / aw;
  t[1]  = (gcy - acy) / ah;
  t[2]  = logf(gw / aw);
  t[3]  = logf(gh / ah);
  t[4]  = (m3[0] - acx) / aw;
  t[5]  = (m3[1] - acy) / ah;
  t[6]  = m3[2] - z0;
  t[7]  = logf(m3[3] / w0);
  t[8]  = logf(m3[4] / h0);
  t[9]  = logf(m3[5] / l0);
  t[10] = m3[6] - ry0;

  // --- smooth-L1 regression losses ---
  float l2sum = 0.0f, l3sum = 0.0f;
  {
    const float4 p = *(const float4*)(b2 + bn * 4);
    const float pa[4] = {p.x, p.y, p.z, p.w};
#pragma unroll
    for (int j = 0; j < 4; ++j) {
      const float tt = (t[j] - means[j]) / stds[j];
      const float d  = pa[j] - tt;
      const float ad = fabsf(d);
      l2sum += (ad < 1.0f) ? 0.5f * d * d : ad - 0.5f;
    }
  }
#pragma unroll
  for (int j = 0; j < 7; ++j) {
    const float tt = (t[4 + j] - means[4 + j]) / stds[4 + j];
    const float d  = b3[bn * 7 + j] - tt;
    const float ad = fabsf(d);
    l3sum += (ad < 1.0f) ? 0.5f * d * d : ad - 0.5f;
  }

  // --- cross entropy (C = 4) ---
  float ce;
  {
    const float4 c = *(const float4*)(cls + bn * 4);
    const float mx  = fmaxf(fmaxf(c.x, c.y), fmaxf(c.z, c.w));
    const float se  = expf(c.x - mx) + expf(c.y - mx) + expf(c.z - mx) + expf(c.w - mx);
    const float lse = mx + logf(se);
    const float sel = (label == 0) ? c.x : (label == 1) ? c.y : (label == 2) ? c.z : c.w;
    ce = lse - sel;
  }

  // --- deterministic block reduction of (ce, fg, l2*fg, l3*fg) ---
  float4 acc;
  acc.x = ce; acc.y = fg; acc.z = l2sum * fg; acc.w = l3sum * fg;
  s_red[tid] = acc;
  __syncthreads();
#pragma unroll
  for (int s = TPB / 2; s > 0; s >>= 1) {
    if (tid < s) {
      float4 o = s_red[tid + s];
      s_red[tid].x += o.x; s_red[tid].y += o.y;
      s_red[tid].z += o.z; s_red[tid].w += o.w;
    }
    __syncthreads();
  }
  if (tid == 0) {
    const int blk = blockIdx.y * gridDim.x + blockIdx.x;
    *(float4*)(partials + (size_t)blk * 4) = s_red[0];
  }
}

__global__ __launch_bounds__(TPB) void rpn_loss_reduce(
    const float* __restrict__ partials, float* __restrict__ out)
{
  __shared__ float4 s[TPB];
  const int tid = threadIdx.x;
  float4 acc = {0.0f, 0.0f, 0.0f, 0.0f};
  for (int i = tid; i < NBLK; i += TPB) {          // fixed stride -> deterministic
    const float4 p = *(const float4*)(partials + (size_t)i * 4);
    acc.x += p.x; acc.y += p.y; acc.z += p.z; acc.w += p.w;
  }
  s[tid] = acc;
  __syncthreads();
#pragma unroll
  for (int st = TPB / 2; st > 0; st >>= 1) {
    if (tid < st) {
      float4 o = s[tid + st];
      s[tid].x += o.x; s[tid].y += o.y; s[tid].z += o.z; s[tid].w += o.w;
    }
    __syncthreads();
  }
  if (tid == 0) {
    const float4 t = s[0];
    const float nfg = fmaxf(t.y, 1.0f);
    out[0] = t.x * (1.0f / ((float)Bb * (float)Nn)) + (t.z + t.w) / nfg;
  }
}

extern "C" void kernel_launch(void* const* d_in, const int* in_sizes, int n_in,
                              void* d_out, int out_size, void* d_ws, size_t ws_size,
                              hipStream_t stream) {
  (void)in_sizes; (void)n_in; (void)out_size; (void)ws_size;
  const float* cls   = (const float*)d_in[0];
  const float* b2    = (const float*)d_in[1];
  const float* b3    = (const float*)d_in[2];
  const float* rois  = (const float*)d_in[3];
  const float* r3    = (const float*)d_in[4];
  const float* g2    = (const float*)d_in[5];
  const float* g3    = (const float*)d_in[6];
  const int*   gc    = (const int*)d_in[7];
  const float* means = (const float*)d_in[8];
  const float* stds  = (const float*)d_in[9];

  float* partials = (float*)d_ws;  // NBLK * 4 floats = 27.6 KB

  dim3 grid(BLK_X, Bb);
  rpn_loss_main<<<grid, TPB, 0, stream>>>(cls, b2, b3, rois, r3, g2, g3, gc,
                                          means, stds, partials);
  rpn_loss_reduce<<<1, TPB, 0, stream>>>(partials, (float*)d_out);
}